// Model_47785806135331
// MI455X (gfx1250) — compile-verified
//
#include <hip/hip_runtime.h>
#include <hip/hip_bf16.h>

typedef __attribute__((ext_vector_type(2))) float v2f;
typedef __attribute__((ext_vector_type(8))) float v8f;

#define N_USER_C  100000
#define N_ITEM_C  200000
#define N_NODE_C  300000
#define LATDIM_C  64

// ---------------------------------------------------------------------------
// Utility: zero / copy (float4 granularity; all sizes are multiples of 4)
// ---------------------------------------------------------------------------
__global__ void zero4_kernel(float4* __restrict__ p, long long n4) {
    long long i = (long long)blockIdx.x * blockDim.x + threadIdx.x;
    if (i < n4) p[i] = make_float4(0.f, 0.f, 0.f, 0.f);
}

__global__ void copy4_kernel(float4* __restrict__ d, const float4* __restrict__ s, long long n4) {
    long long i = (long long)blockIdx.x * blockDim.x + threadIdx.x;
    if (i < n4) d[i] = s[i];
}

// ---------------------------------------------------------------------------
// SpMM: y[row] += val * x[col]  (rows random -> atomic accumulate).
// 16 lanes per edge, float4 per lane (16*4 = 64 dims). Accumulator matrix is
// <= 76.8 MB -> L2 resident on MI455X, so atomics resolve on-chip.
// ---------------------------------------------------------------------------
__global__ void spmm_kernel(const int* __restrict__ rows, const int* __restrict__ cols,
                            const float* __restrict__ vals, const float* __restrict__ x,
                            float* y, int nedges) {
    long long t = (long long)blockIdx.x * blockDim.x + threadIdx.x;
    long long e = t >> 4;
    int l = (int)(t & 15);
    if (e >= nedges) return;
    int r = rows[e];
    int c = cols[e];
    float v = vals[e];
    float4 xv = ((const float4*)(x + (long long)c * LATDIM_C))[l];
    float* yp = y + (long long)r * LATDIM_C + l * 4;
    unsafeAtomicAdd(yp + 0, v * xv.x);
    unsafeAtomicAdd(yp + 1, v * xv.y);
    unsafeAtomicAdd(yp + 2, v * xv.z);
    unsafeAtomicAdd(yp + 3, v * xv.w);
}

// ---------------------------------------------------------------------------
// Gate + fuse (per layer k):
//   logits = [uu|ui] @ W^T + b  (W: 2x128)  -> leaky_relu -> softmax(2)
//   out_row = m0*uu + m1*ui     (writes user rows of `outp`, item rows untouched)
// One wave handles 16 users. Logit GEMM done with V_WMMA_F32_16X16X4_F32:
//   A (16x4 f32): lane = M + 16*(K/2), vgpr = K%2
//   B (4x16 f32): lane = N + 16*(K/2), vgpr = K%2
//   C/D (16x16) : vgpr r, lanes 0-15 -> M=r, lanes 16-31 -> M=r+8, N=lane%16
// B operand: address-clamped unconditional float2 load + v_cndmask zeroing of
// dead columns (no divergent loads / EXEC save-restore in the WMMA loop).
// May run in place (ui == outp): each wave reads its rows before writing them.
// ---------------------------------------------------------------------------
__global__ __launch_bounds__(256) void gate_fuse_kernel(
    const float* __restrict__ uu,   // N_USER x 64 (soc[k])
    const float* ui,                // N_USER x 64 (inter user part; may alias outp)
    const float* __restrict__ Wg,   // 2 x 128
    const float* __restrict__ bg,   // 2
    float* outp,                    // >= N_USER x 64
    int nrows) {
    int lane = threadIdx.x & 31;
    long long u0 = ((long long)blockIdx.x * 8 + (threadIdx.x >> 5)) * 16;
    if (u0 >= nrows) return;               // wave-uniform -> EXEC all-1s for WMMA
    int m = lane & 15;                     // A row / B column handled by this lane
    int ksub = (lane >> 4) * 2;            // K sub-offset for this lane half
    const float* rowu = uu + (u0 + m) * (long long)LATDIM_C;
    const float* rowi = ui + (u0 + m) * (long long)LATDIM_C;
    const bool act = (m < 2);
    const float* wrow = Wg + (act ? m : 1) * 128;   // clamped row: always in-bounds
    v8f c = {};
#pragma unroll
    for (int kk = 0; kk < 128; kk += 4) {
        const float* src = (kk < 64) ? rowu : rowi;   // compile-time per iteration
        int jj = (kk & 63) + ksub;
        v2f a, b;
        a[0] = src[jj];
        a[1] = src[jj + 1];
        float2 wv = *(const float2*)(wrow + kk + ksub);  // unconditional, 8B-aligned
        b[0] = act ? wv.x : 0.f;
        b[1] = act ? wv.y : 0.f;
        c = __builtin_amdgcn_wmma_f32_16x16x4_f32(false, a, false, b, (short)0, c, false, false);
    }
    float b0 = bg[0], b1 = bg[1];
#pragma unroll
    for (int u = 0; u < 16; ++u) {
        int half = (u >> 3) * 16;
        float g0 = __shfl(c[u & 7], half + 0, 32) + b0;
        float g1 = __shfl(c[u & 7], half + 1, 32) + b1;
        g0 = g0 > 0.f ? g0 : 0.01f * g0;
        g1 = g1 > 0.f ? g1 : 0.01f * g1;
        float mx = fmaxf(g0, g1);
        float e0 = __expf(g0 - mx), e1 = __expf(g1 - mx);
        float inv = 1.f / (e0 + e1);
        float m0 = e0 * inv, m1 = e1 * inv;
        long long row = u0 + u;
        float2 au = ((const float2*)(uu + row * LATDIM_C))[lane];
        float2 ai = ((const float2*)(ui + row * LATDIM_C))[lane];
        float2 o;
        o.x = au.x * m0 + ai.x * m1;
        o.y = au.y * m0 + ai.y * m1;
        ((float2*)(outp + row * LATDIM_C))[lane] = o;
    }
}

// ---------------------------------------------------------------------------
// Final attention fuse:
//   cu = [f0|f1|f2|f3|f4] (rowbase..rowbase+nrows), logits = cu @ W^T + b
//   (W: 5x320) -> leaky_relu -> softmax(5) -> out_row = sum_k w_k * fk_row
// Same WMMA scheme, K = 320 (80 steps), 5 live logit columns.
// ---------------------------------------------------------------------------
__global__ __launch_bounds__(256) void attn_fuse_kernel(
    const float* __restrict__ f0, const float* __restrict__ f1,
    const float* __restrict__ f2, const float* __restrict__ f3,
    const float* __restrict__ f4,
    const float* __restrict__ W,    // 5 x 320
    const float* __restrict__ bias, // 5
    float* __restrict__ outp,       // N_NODE x 64 (writes rows rowbase..)
    long long rowbase, int nrows) {
    int lane = threadIdx.x & 31;
    long long u0 = ((long long)blockIdx.x * 8 + (threadIdx.x >> 5)) * 16;
    if (u0 >= nrows) return;
    int m = lane & 15;
    int ksub = (lane >> 4) * 2;
    long long row_m = rowbase + u0 + m;
    const float* fr[5] = {
        f0 + row_m * LATDIM_C, f1 + row_m * LATDIM_C, f2 + row_m * LATDIM_C,
        f3 + row_m * LATDIM_C, f4 + row_m * LATDIM_C };
    const bool act = (m < 5);
    const float* wrow = W + (act ? m : 4) * 320;    // clamped row: always in-bounds
    v8f c = {};
#pragma unroll
    for (int kk = 0; kk < 320; kk += 4) {
        const float* src = fr[kk >> 6];     // compile-time select after unroll
        int jj = (kk & 63) + ksub;
        v2f a, b;
        a[0] = src[jj];
        a[1] = src[jj + 1];
        float2 wv = *(const float2*)(wrow + kk + ksub);  // unconditional, 8B-aligned
        b[0] = act ? wv.x : 0.f;
        b[1] = act ? wv.y : 0.f;
        c = __builtin_amdgcn_wmma_f32_16x16x4_f32(false, a, false, b, (short)0, c, false, false);
    }
    float bb0 = bias[0], bb1 = bias[1], bb2 = bias[2], bb3 = bias[3], bb4 = bias[4];
    const float* fsrc[5] = { f0, f1, f2, f3, f4 };
#pragma unroll
    for (int u = 0; u < 16; ++u) {
        int half = (u >> 3) * 16;
        float g[5];
        g[0] = __shfl(c[u & 7], half + 0, 32) + bb0;
        g[1] = __shfl(c[u & 7], half + 1, 32) + bb1;
        g[2] = __shfl(c[u & 7], half + 2, 32) + bb2;
        g[3] = __shfl(c[u & 7], half + 3, 32) + bb3;
        g[4] = __shfl(c[u & 7], half + 4, 32) + bb4;
        float mx = g[0];
#pragma unroll
        for (int n = 0; n < 5; ++n) {
            g[n] = g[n] > 0.f ? g[n] : 0.01f * g[n];
            mx = fmaxf(mx, g[n]);
        }
        float e[5], sum = 0.f;
#pragma unroll
        for (int n = 0; n < 5; ++n) { e[n] = __expf(g[n] - mx); sum += e[n]; }
        float inv = 1.f / sum;
        long long row = rowbase + u0 + u;
        float2 acc = make_float2(0.f, 0.f);
#pragma unroll
        for (int n = 0; n < 5; ++n) {
            float w = e[n] * inv;
            float2 v = ((const float2*)(fsrc[n] + row * LATDIM_C))[lane];
            acc.x += w * v.x;
            acc.y += w * v.y;
        }
        ((float2*)(outp + row * LATDIM_C))[lane] = acc;
    }
}

// ---------------------------------------------------------------------------
extern "C" void kernel_launch(void* const* d_in, const int* in_sizes, int n_in,
                              void* d_out, int out_size, void* d_ws, size_t ws_size,
                              hipStream_t stream) {
    const float* uE  = (const float*)d_in[0];   // 100000 x 64
    const float* iE  = (const float*)d_in[1];   // 200000 x 64
    const float* Wg  = (const float*)d_in[2];   // 5 x 2 x 128
    const float* bg  = (const float*)d_in[3];   // 5 x 2
    const float* WL1 = (const float*)d_in[4];   // 5 x 320
    const float* bL1 = (const float*)d_in[5];   // 5
    const float* WL2 = (const float*)d_in[6];   // 5 x 320
    const float* bL2 = (const float*)d_in[7];   // 5
    const int*   ir  = (const int*)d_in[8];     // E_INTER
    const int*   ic  = (const int*)d_in[9];
    const float* iv  = (const float*)d_in[10];
    const int*   sr  = (const int*)d_in[11];    // E_SOC
    const int*   sc  = (const int*)d_in[12];
    const float* sv  = (const float*)d_in[13];
    float* out = (float*)d_out;

    const int E_INTER = in_sizes[8];
    const int E_SOC   = in_sizes[11];

    // workspace layout: 5 fused node matrices + 4 soc user matrices
    const size_t PF = (size_t)N_NODE_C * LATDIM_C;   // 19.2M floats
    const size_t PS = (size_t)N_USER_C * LATDIM_C;   //  6.4M floats
    float* ws = (float*)d_ws;
    float* f[5];
    for (int k = 0; k < 5; ++k) f[k] = ws + (size_t)k * PF;
    float* s[5];
    s[0] = nullptr; // = uE
    for (int t = 1; t <= 4; ++t) s[t] = ws + 5 * PF + (size_t)(t - 1) * PS;

    const int TPB = 256;
    const int gate_blocks = (int)(((long long)N_USER_C / 16 + 7) / 8);          // 782
    const int attn_u_blocks = gate_blocks;
    const int attn_i_blocks = (int)(((long long)N_ITEM_C / 16 + 7) / 8);        // 1563
    const int zf_blocks = (int)((PF / 4 + TPB - 1) / TPB);
    const int zs_blocks = (int)((PS / 4 + TPB - 1) / TPB);
    const int spmm_i_blocks = (int)(((long long)E_INTER * 16 + TPB - 1) / TPB);
    const int spmm_s_blocks = (int)(((long long)E_SOC * 16 + TPB - 1) / TPB);

    // 1) social propagation: s[t] = spmm(soc, s[t-1]), s[0] = uE
    for (int t = 1; t <= 4; ++t) {
        zero4_kernel<<<zs_blocks, TPB, 0, stream>>>((float4*)s[t], (long long)(PS / 4));
        const float* xsrc = (t == 1) ? uE : s[t - 1];
        spmm_kernel<<<spmm_s_blocks, TPB, 0, stream>>>(sr, sc, sv, xsrc, s[t], E_SOC);
    }

    // 2) layer 0: fused[0] = [gate(uE,uE) user rows | iE item rows]
    copy4_kernel<<<(int)(((size_t)N_ITEM_C * LATDIM_C / 4 + TPB - 1) / TPB), TPB, 0, stream>>>(
        (float4*)(f[0] + (size_t)N_USER_C * LATDIM_C), (const float4*)iE,
        (long long)((size_t)N_ITEM_C * LATDIM_C / 4));
    gate_fuse_kernel<<<gate_blocks, TPB, 0, stream>>>(uE, uE, Wg, bg, f[0], N_USER_C);
    zero4_kernel<<<zf_blocks, TPB, 0, stream>>>((float4*)f[1], (long long)(PF / 4));
    spmm_kernel<<<spmm_i_blocks, TPB, 0, stream>>>(ir, ic, iv, f[0], f[1], E_INTER);

    // 3) layers 1..4: in-place gate on user rows of f[k]; spmm -> f[k+1]
    for (int k = 1; k <= 4; ++k) {
        gate_fuse_kernel<<<gate_blocks, TPB, 0, stream>>>(
            s[k], f[k], Wg + (size_t)k * 2 * 128, bg + (size_t)k * 2, f[k], N_USER_C);
        if (k < 4) {
            zero4_kernel<<<zf_blocks, TPB, 0, stream>>>((float4*)f[k + 1], (long long)(PF / 4));
            spmm_kernel<<<spmm_i_blocks, TPB, 0, stream>>>(ir, ic, iv, f[k], f[k + 1], E_INTER);
        }
        // note: the reference's k==4 interaction spmm output is never used -> skipped
    }

    // 4) final attention fusion -> output
    attn_fuse_kernel<<<attn_u_blocks, TPB, 0, stream>>>(
        f[0], f[1], f[2], f[3], f[4], WL1, bL1, out, 0LL, N_USER_C);
    attn_fuse_kernel<<<attn_i_blocks, TPB, 0, stream>>>(
        f[0], f[1], f[2], f[3], f[4], WL2, bL2, out, (long long)N_USER_C, N_ITEM_C);
}